// AttentionLoss_30133490549307
// MI455X (gfx1250) — compile-verified
//
#include <hip/hip_runtime.h>
#include <math.h>

// ---------------------------------------------------------------------------
// AttentionLoss for MI455X (gfx1250, wave32).
// Memory-bound reduction pipeline; block reductions use V_WMMA_F32_16X16X4_F32
// (rowsum-via-ones-matrix, mathematically exact). Channel-strided passes use
// b128 loads (512B/wave/instr) + global_prefetch for the 16KB-strided rows.
// ---------------------------------------------------------------------------

typedef float v2f __attribute__((ext_vector_type(2)));
typedef float v8f __attribute__((ext_vector_type(8)));

#define BLOCK 256
#define BS_ 8
#define A_COEF_ 4e-4f
#define B_COEF_ 2e-2f
#define INV_T_ 2.0f   // 1/T, T = 0.5

// Sum-reduce 256 per-thread floats. Wave 0 interprets the 256 LDS values as a
// 16x16 f32 matrix and computes row sums with 4 chained V_WMMA_F32_16X16X4_F32
// (B = all-ones: layout-independent). Result broadcast to all threads.
__device__ __forceinline__ float blockReduceSum256(float val, float* lds) {
  const int tid = threadIdx.x;
  __syncthreads();              // protect LDS reuse across consecutive calls
  lds[tid] = val;
  __syncthreads();
  if (tid < 32) {               // whole wave 0 => EXEC all ones (WMMA req.)
#if __has_builtin(__builtin_amdgcn_wmma_f32_16x16x4_f32)
    const int row = tid & 15;
    const int kb  = (tid >> 4) << 1;   // lanes 0-15: K={0,1}; 16-31: K={2,3}
    v8f c = {};
    v2f ones; ones[0] = 1.0f; ones[1] = 1.0f;
#pragma unroll
    for (int j = 0; j < 4; ++j) {      // K-chunks of 4 columns
      v2f a;
      a[0] = lds[row * 16 + 4 * j + kb + 0];
      a[1] = lds[row * 16 + 4 * j + kb + 1];
      c = __builtin_amdgcn_wmma_f32_16x16x4_f32(false, a, false, ones,
                                                (short)0, c, false, false);
    }
    // vgpr r, lanes 0-15 hold rowsum(M=r); lanes 16-31 hold rowsum(M=8+r)
    float t = ((c[0] + c[1]) + (c[2] + c[3])) + ((c[4] + c[5]) + (c[6] + c[7]));
    t += __shfl_xor(t, 16, 32);        // rows 0-7 + rows 8-15
    if (tid == 0) lds[0] = t;
#else
    float t = lds[tid];
#pragma unroll
    for (int i = 1; i < 8; ++i) t += lds[tid + 32 * i];
#pragma unroll
    for (int m = 16; m > 0; m >>= 1) t += __shfl_xor(t, m, 32);
    if (tid == 0) lds[0] = t;
#endif
  }
  __syncthreads();
  return lds[0];
}

__device__ __forceinline__ float blockReduceMax256(float val, float* lds) {
  const int tid = threadIdx.x;
  __syncthreads();
  lds[tid] = val;
  __syncthreads();
  for (int s = 128; s > 0; s >>= 1) {
    if (tid < s) lds[tid] = fmaxf(lds[tid], lds[tid + s]);
    __syncthreads();
  }
  return lds[0];
}

// ---------------------------------------------------------------------------
__global__ void zeroKernel(float* p, int n) {
  int i = blockIdx.x * BLOCK + threadIdx.x;
  if (i < n) p[i] = 0.0f;
}

// One block per (b*C + c): raw channel sums of |s|, |t| over contiguous HW.
__global__ void gcKernel(const float* __restrict__ s, const float* __restrict__ t,
                         float* __restrict__ gcS, float* __restrict__ gcT, int HW) {
  __shared__ float lds[BLOCK];
  const int tid = threadIdx.x;
  const size_t base = (size_t)blockIdx.x * (size_t)HW;
  const float4* s4 = (const float4*)(s + base);
  const float4* t4 = (const float4*)(t + base);
  const int n4 = HW >> 2;
  float as = 0.0f, at = 0.0f;
  for (int i = tid; i < n4; i += BLOCK) {
    float4 v = s4[i];
    as += (fabsf(v.x) + fabsf(v.y)) + (fabsf(v.z) + fabsf(v.w));
    float4 w = t4[i];
    at += (fabsf(w.x) + fabsf(w.y)) + (fabsf(w.z) + fabsf(w.w));
  }
  float ss = blockReduceSum256(as, lds);
  float st = blockReduceSum256(at, lds);
  if (tid == 0) { gcS[blockIdx.x] = ss; gcT[blockIdx.x] = st; }
}

// Thread owns 4 consecutive hw (b128 loads), loops a channel sub-range
// (CSPLIT blocks cooperate via fp32 atomics) -> raw spatial sums over C.
__global__ void gsKernel(const float* __restrict__ s, const float* __restrict__ t,
                         float* __restrict__ gsS, float* __restrict__ gsT,
                         int C, int HW, int CSPLIT) {
  const int tid   = threadIdx.x;
  const int bpb   = HW / (BLOCK * 4);      // hw-blocks per (b, cs)
  const int hwblk = blockIdx.x % bpb;
  const int rest  = blockIdx.x / bpb;
  const int cs    = rest % CSPLIT;
  const int b     = rest / CSPLIT;
  const int cPer  = C / CSPLIT;
  const int hw4   = hwblk * BLOCK + tid;   // float4 index within row
  const int str4  = HW >> 2;               // row stride in float4
  const float4* sp = (const float4*)(s + (size_t)b * C * HW + (size_t)(cs * cPer) * HW) + hw4;
  const float4* tp = (const float4*)(t + (size_t)b * C * HW + (size_t)(cs * cPer) * HW) + hw4;
  float ax = 0.f, ay = 0.f, az = 0.f, aw = 0.f;
  float bx = 0.f, by = 0.f, bz = 0.f, bw = 0.f;
#pragma unroll 4
  for (int c = 0; c < cPer; ++c) {
    __builtin_prefetch(sp + (size_t)(c + 8) * str4, 0, 3);
    __builtin_prefetch(tp + (size_t)(c + 8) * str4, 0, 3);
    float4 v = sp[(size_t)c * str4];
    ax += fabsf(v.x); ay += fabsf(v.y); az += fabsf(v.z); aw += fabsf(v.w);
    float4 w = tp[(size_t)c * str4];
    bx += fabsf(w.x); by += fabsf(w.y); bz += fabsf(w.z); bw += fabsf(w.w);
  }
  float* ds = &gsS[b * HW + hw4 * 4];
  float* dt = &gsT[b * HW + hw4 * 4];
  atomicAdd(ds + 0, ax); atomicAdd(ds + 1, ay); atomicAdd(ds + 2, az); atomicAdd(ds + 3, aw);
  atomicAdd(dt + 0, bx); atomicAdd(dt + 1, by); atomicAdd(dt + 2, bz); atomicAdd(dt + 3, bw);
}

// Per-batch block: mask = N * softmax((a+b)/T) with a = rawA*invDiv, and
// accumulate MSE(a,b) into lat_acc. Used for both spatial (N=HW) and channel
// (N=C) masks.
__global__ void softmaxMaskKernel(const float* __restrict__ gA, const float* __restrict__ gB,
                                  float* __restrict__ mask, int N, float invDiv,
                                  float* __restrict__ latAcc) {
  __shared__ float lds[BLOCK];
  const int tid = threadIdx.x;
  const int b   = blockIdx.x;
  const float* pa = gA + (size_t)b * N;
  const float* pb = gB + (size_t)b * N;
  float lmax = -INFINITY, lat = 0.0f;
  for (int i = tid; i < N; i += BLOCK) {
    float a = pa[i] * invDiv, c = pb[i] * invDiv;
    float d = a - c;
    lat += d * d;
    lmax = fmaxf(lmax, (a + c) * INV_T_);
  }
  const float gmax = blockReduceMax256(lmax, lds);
  float lsum = 0.0f;
  for (int i = tid; i < N; i += BLOCK)
    lsum += expf((pa[i] + pb[i]) * invDiv * INV_T_ - gmax);
  const float gsum  = blockReduceSum256(lsum, lds);
  const float scale = (float)N / gsum;
  for (int i = tid; i < N; i += BLOCK)
    mask[(size_t)b * N + i] = expf((pa[i] + pb[i]) * invDiv * INV_T_ - gmax) * scale;
  const float latTot = blockReduceSum256(lat, lds);
  if (tid == 0) atomicAdd(latAcc, latTot);
}

// slam partial: thread owns 4 hw (b128 loads), loops channel sub-range;
// Mc staged in LDS; partial = sum_hw Ms[hw] * sum_c (s-t)^2 * Mc[c];
// WMMA block reduce -> one atomic per block.
__global__ void slamKernel(const float* __restrict__ s, const float* __restrict__ t,
                           const float* __restrict__ Ms, const float* __restrict__ Mc,
                           int C, int HW, int CSPLIT, float* __restrict__ slamAcc) {
  __shared__ float ldsRed[BLOCK];
  __shared__ float ldsMc[64];
  const int tid   = threadIdx.x;
  const int bpb   = HW / (BLOCK * 4);
  const int hwblk = blockIdx.x % bpb;
  const int rest  = blockIdx.x / bpb;
  const int cs    = rest % CSPLIT;
  const int b     = rest / CSPLIT;
  const int cPer  = C / CSPLIT;            // 16 for both levels
  const int c0    = cs * cPer;
  const int hw4   = hwblk * BLOCK + tid;
  const int str4  = HW >> 2;
  for (int i = tid; i < cPer; i += BLOCK) ldsMc[i] = Mc[b * C + c0 + i];
  __syncthreads();
  const float4 ms = ((const float4*)(Ms + (size_t)b * HW))[hw4];
  const float4* sp = (const float4*)(s + (size_t)b * C * HW + (size_t)c0 * HW) + hw4;
  const float4* tp = (const float4*)(t + (size_t)b * C * HW + (size_t)c0 * HW) + hw4;
  float ax = 0.f, ay = 0.f, az = 0.f, aw = 0.f;
#pragma unroll 4
  for (int c = 0; c < cPer; ++c) {
    __builtin_prefetch(sp + (size_t)(c + 8) * str4, 0, 3);
    __builtin_prefetch(tp + (size_t)(c + 8) * str4, 0, 3);
    float4 v = sp[(size_t)c * str4];
    float4 w = tp[(size_t)c * str4];
    float m = ldsMc[c];
    float dx = v.x - w.x, dy = v.y - w.y, dz = v.z - w.z, dw = v.w - w.w;
    ax = fmaf(dx * dx, m, ax);
    ay = fmaf(dy * dy, m, ay);
    az = fmaf(dz * dz, m, az);
    aw = fmaf(dw * dw, m, aw);
  }
  const float partial = fmaf(ax, ms.x, fmaf(ay, ms.y, fmaf(az, ms.z, aw * ms.w)));
  const float total = blockReduceSum256(partial, ldsRed);
  if (tid == 0) atomicAdd(slamAcc, total);
}

// acc: [0]=lat, [1]=slam level1, [2]=slam level2
__global__ void finalizeKernel(const float* __restrict__ acc, float* __restrict__ out) {
  if (threadIdx.x == 0 && blockIdx.x == 0) {
    float lat = acc[0];
    float lam = sqrtf(acc[1]) + sqrtf(acc[2]);
    float att = (A_COEF_ * lat + B_COEF_ * lam) / ((float)BS_ * 2.0f);
    out[0] = att * (float)BS_;
    out[1] = att;
  }
}

// ---------------------------------------------------------------------------
extern "C" void kernel_launch(void* const* d_in, const int* in_sizes, int n_in,
                              void* d_out, int out_size, void* d_ws, size_t ws_size,
                              hipStream_t stream) {
  const float* stu1 = (const float*)d_in[1];
  const float* tea1 = (const float*)d_in[2];
  const float* stu2 = (const float*)d_in[3];
  const float* tea2 = (const float*)d_in[4];
  float* out = (float*)d_out;

  const int B = 8;
  const int C1 = 256, HW1 = 64 * 64;   // 4096
  const int C2 = 512, HW2 = 32 * 32;   // 1024
  const int CS1 = 16, CS2 = 32;        // channel splits (cPer = 16)

  // Workspace layout (floats). Zeroed region first (Gs sums via atomics + acc).
  float* w = (float*)d_ws;
  float* gsS1 = w;               w += (size_t)B * HW1;   // 32768
  float* gsT1 = w;               w += (size_t)B * HW1;
  float* gsS2 = w;               w += (size_t)B * HW2;   // 8192
  float* gsT2 = w;               w += (size_t)B * HW2;
  float* acc  = w;               w += 4;                 // lat, slam1, slam2
  const int nZero = 2 * B * HW1 + 2 * B * HW2 + 4;
  float* gcS1 = w;               w += (size_t)B * C1;
  float* gcT1 = w;               w += (size_t)B * C1;
  float* gcS2 = w;               w += (size_t)B * C2;
  float* gcT2 = w;               w += (size_t)B * C2;
  float* Ms1  = w;               w += (size_t)B * HW1;
  float* Mc1  = w;               w += (size_t)B * C1;
  float* Ms2  = w;               w += (size_t)B * HW2;
  float* Mc2  = w;               w += (size_t)B * C2;

  // 1) zero atomically-accumulated buffers (every launch: graph replays)
  zeroKernel<<<(nZero + BLOCK - 1) / BLOCK, BLOCK, 0, stream>>>((float*)d_ws, nZero);

  // 2) channel sums (contiguous HW reductions, one block per (b,c))
  gcKernel<<<B * C1, BLOCK, 0, stream>>>(stu1, tea1, gcS1, gcT1, HW1);
  gcKernel<<<B * C2, BLOCK, 0, stream>>>(stu2, tea2, gcS2, gcT2, HW2);

  // 3) spatial sums (b128 strided over C, c-split + atomics)
  gsKernel<<<B * CS1 * (HW1 / (BLOCK * 4)), BLOCK, 0, stream>>>(stu1, tea1, gsS1, gsT1, C1, HW1, CS1);
  gsKernel<<<B * CS2 * (HW2 / (BLOCK * 4)), BLOCK, 0, stream>>>(stu2, tea2, gsS2, gsT2, C2, HW2, CS2);

  // 4) softmax masks + lat MSE (tiny). invDiv: spatial=1/C, channel=1/HW.
  softmaxMaskKernel<<<B, BLOCK, 0, stream>>>(gsS1, gsT1, Ms1, HW1, 1.0f / C1,  acc);
  softmaxMaskKernel<<<B, BLOCK, 0, stream>>>(gcS1, gcT1, Mc1, C1,  1.0f / HW1, acc);
  softmaxMaskKernel<<<B, BLOCK, 0, stream>>>(gsS2, gsT2, Ms2, HW2, 1.0f / C2,  acc);
  softmaxMaskKernel<<<B, BLOCK, 0, stream>>>(gcS2, gcT2, Mc2, C2,  1.0f / HW2, acc);

  // 5) weighted MSE (second pass; working set fits 192MB L2)
  slamKernel<<<B * CS1 * (HW1 / (BLOCK * 4)), BLOCK, 0, stream>>>(stu1, tea1, Ms1, Mc1, C1, HW1, CS1, acc + 1);
  slamKernel<<<B * CS2 * (HW2 / (BLOCK * 4)), BLOCK, 0, stream>>>(stu2, tea2, Ms2, Mc2, C2, HW2, CS2, acc + 2);

  // 6) combine
  finalizeKernel<<<1, 32, 0, stream>>>(acc, out);

  (void)in_sizes; (void)n_in; (void)out_size; (void)ws_size;
}